// FieldAwareFMLayer_35450660061570
// MI455X (gfx1250) — compile-verified
//
#include <hip/hip_runtime.h>
#include <stdint.h>

// FFM pairwise interaction reduction for MI455X (gfx1250, wave32).
// out[b] = sum_{f1<f2} dot(x[b,f1,f2-1,:], x[b,f2,f1,:]),  x: (bs, 40, 39, 16) f32.
// Memory-bound: 204.5 MB streamed once -> ~8.8 us floor at 23.3 TB/s.

#define NF        40
#define NPAIRS    780                    // 40*39/2
#define ROW_ELEMS (NF * (NF - 1) * 16)   // 24960 floats per batch row
#define BT        256                    // 8 wave32 per block

typedef __attribute__((ext_vector_type(2))) float v2f;
typedef __attribute__((ext_vector_type(8))) float v8f;

// Compile-time pair table: entry p packs (emb0_off << 16) | emb1_off,
// offsets in float elements (max 24944 < 65536). Replaces all runtime
// triangular-index inversion math with one coalesced 4B load per pair.
struct PairTab { uint32_t v[NPAIRS]; };
static constexpr PairTab make_pair_tab() {
    PairTab t{};
    int p = 0;
    for (int i = 0; i < NF - 1; ++i) {
        for (int j = i; j < NF - 1; ++j) {
            const uint32_t off0 = (uint32_t)((i * (NF - 1) + j) * 16);        // x[i][j]
            const uint32_t off1 = (uint32_t)(((j + 1) * (NF - 1) + i) * 16);  // x[j+1][i]
            t.v[p++] = (off0 << 16) | off1;
        }
    }
    return t;
}
__constant__ PairTab PAIR_TAB = make_pair_tab();

__global__ __launch_bounds__(BT) void ffm_interact_kernel(
    const float* __restrict__ in, float* __restrict__ out)
{
    // 16 KB staging tile: 64 B per thread per iteration, filled by the async engine.
    __shared__ __align__(16) float tile[BT * 16];
    __shared__ float wsum[BT / 32];

    const int tid = threadIdx.x;
    const int bid = blockIdx.x;
    const float* row = in + (size_t)bid * ROW_ELEMS;

    // LDS byte address of this thread's slice (low 32 bits of the generic
    // pointer are the group-segment byte offset on CDNA5).
    const uint32_t ldsb = (uint32_t)(uintptr_t)(&tile[tid * 16]);

    float4 acc = make_float4(0.f, 0.f, 0.f, 0.f);

    for (int p0 = 0; p0 < NPAIRS; p0 += BT) {
        // Previous iteration's LDS reads must retire before the async engine
        // overwrites this thread's slice.
        asm volatile("s_wait_dscnt 0x0" ::: "memory");

        const int  p   = p0 + tid;
        const bool act = (p < NPAIRS);
        const int  pc  = act ? p : (NPAIRS - 1);

        const uint32_t pk   = PAIR_TAB.v[pc];
        const uint32_t off0 = pk >> 16;       // emb0 = x[i][j]      (linear stream)
        const uint32_t off1 = pk & 0xFFFFu;   // emb1 = x[j+1][i]    (gathered)

        float4 b0 = make_float4(0.f, 0.f, 0.f, 0.f), b1 = b0, b2 = b0, b3 = b0;
        if (act) {
            const float* g0 = row + off0;
            // CDNA5 async global->LDS copy; inst offset advances BOTH the LDS
            // and the global address (CDNA5 ISA), so four b128s
            // stage the contiguous 64 B emb0 chunk. Tracked by ASYNCcnt.
            asm volatile(
                "global_load_async_to_lds_b128 %0, %1, off\n\t"
                "global_load_async_to_lds_b128 %0, %1, off offset:16\n\t"
                "global_load_async_to_lds_b128 %0, %1, off offset:32\n\t"
                "global_load_async_to_lds_b128 %0, %1, off offset:48"
                :
                : "v"(ldsb), "v"(g0)
                : "memory");
            // Gathered emb1 side goes straight to VGPRs (64 B contiguous per lane).
            const float4* g1 = (const float4*)(row + off1);
            b0 = g1[0]; b1 = g1[1]; b2 = g1[2]; b3 = g1[3];
        }

        asm volatile("s_wait_asynccnt 0x0" ::: "memory");

        if (act) {
            const float4* t4 = (const float4*)(&tile[tid * 16]);
            float4 a0 = t4[0], a1 = t4[1], a2 = t4[2], a3 = t4[3];
            acc.x += a0.x * b0.x; acc.y += a0.y * b0.y;
            acc.z += a0.z * b0.z; acc.w += a0.w * b0.w;
            acc.x += a1.x * b1.x; acc.y += a1.y * b1.y;
            acc.z += a1.z * b1.z; acc.w += a1.w * b1.w;
            acc.x += a2.x * b2.x; acc.y += a2.y * b2.y;
            acc.z += a2.z * b2.z; acc.w += a2.w * b2.w;
            acc.x += a3.x * b3.x; acc.y += a3.y * b3.y;
            acc.z += a3.z * b3.z; acc.w += a3.w * b3.w;
        }
    }

    // Wave32 reduction.
    float a = acc.x + acc.y + acc.z + acc.w;
#pragma unroll
    for (int m = 16; m >= 1; m >>= 1) a += __shfl_xor(a, m, 32);
    if ((tid & 31) == 0) wsum[tid >> 5] = a;
    __syncthreads();

    // Final 8-partial reduction through the matrix pipe (one WMMA per block):
    // A rows 0..1 carry the partials, B is a ones-column, D[0,0]+D[1,0] = total.
    if (tid < 32) {  // wave-uniform branch -> EXEC all ones at the WMMA
        const int lane = tid;
#if __has_builtin(__builtin_amdgcn_wmma_f32_16x16x4_f32)
        v2f A = {0.f, 0.f}, B = {0.f, 0.f};
        // f32 16x4 A layout: lane m -> (M=m, K0|K1), lane m+16 -> (M=m, K2|K3).
        if (lane == 0)       { A.x = wsum[0]; A.y = wsum[1]; B.x = 1.f; B.y = 1.f; }
        else if (lane == 1)  { A.x = wsum[4]; A.y = wsum[5]; }
        else if (lane == 16) { A.x = wsum[2]; A.y = wsum[3]; B.x = 1.f; B.y = 1.f; }
        else if (lane == 17) { A.x = wsum[6]; A.y = wsum[7]; }
        v8f C = {};
        C = __builtin_amdgcn_wmma_f32_16x16x4_f32(
            /*neg_a=*/false, A, /*neg_b=*/false, B,
            /*c_mod=*/(short)0, C, /*reuse_a=*/false, /*reuse_b=*/false);
        if (lane == 0) out[bid] = C[0] + C[1];   // D[0,0] + D[1,0]
#else
        if (lane == 0) {
            float s = 0.f;
#pragma unroll
            for (int k = 0; k < BT / 32; ++k) s += wsum[k];
            out[bid] = s;
        }
#endif
    }
}

extern "C" void kernel_launch(void* const* d_in, const int* in_sizes, int n_in,
                              void* d_out, int out_size, void* d_ws, size_t ws_size,
                              hipStream_t stream)
{
    (void)n_in; (void)out_size; (void)d_ws; (void)ws_size;
    const float* in  = (const float*)d_in[0];
    float*       out = (float*)d_out;
    const int bs = in_sizes[0] / ROW_ELEMS;   // 2048 for the reference setup
    if (bs <= 0) return;
    ffm_interact_kernel<<<dim3(bs), dim3(BT), 0, stream>>>(in, out);
}